// TripletLoss_65712999629575
// MI455X (gfx1250) — compile-verified
//
#include <hip/hip_runtime.h>
#include <hip/hip_bf16.h>
#include <math.h>

typedef __attribute__((ext_vector_type(2))) float v2f;
typedef __attribute__((ext_vector_type(4))) float v4f;
typedef __attribute__((ext_vector_type(8))) float v8f;
typedef __attribute__((ext_vector_type(4))) int   v4i;

// Address-space-qualified int4 for the async-to-LDS builtin
// (prototype per hipcc diagnostic: src is int4* in global AS, dst in LDS AS).
typedef __attribute__((address_space(1))) v4i gas_v4i;
typedef __attribute__((address_space(3))) v4i las_v4i;

#define EPS_D 1e-12f
#define NW 4            // waves per workgroup in hardmine
#define AS_STRIDE 516   // 512 + 4 pad: kills 64-bank conflicts for half-wave row stride

// ---------------- Kernel A: row squared norms ----------------
__global__ void rownorm_kernel(const float* __restrict__ X, float* __restrict__ sq,
                               int n, int d) {
    int wave = threadIdx.x >> 5;
    int lane = threadIdx.x & 31;
    int row = blockIdx.x * (blockDim.x >> 5) + wave;
    if (row >= n) return;
    const float* xr = X + (size_t)row * d;
    float s = 0.f;
    for (int t = lane; t < d; t += 32) { float v = xr[t]; s += v * v; }
    for (int off = 16; off > 0; off >>= 1) s += __shfl_down(s, off, 32);
    if (lane == 0) sq[row] = s;
}

// ---------------- Kernel B: fused WMMA Gram + distance + hard mining ----------------
__global__ __launch_bounds__(NW * 32)
void hardmine_kernel(const float* __restrict__ X, const int* __restrict__ tgt,
                     const float* __restrict__ sq,
                     float* __restrict__ dist_ap, float* __restrict__ dist_an,
                     int* __restrict__ jstar, int* __restrict__ kstar,
                     int n, int d) {
    __shared__ float As[16 * AS_STRIDE];          // 33 KB A panel
    __shared__ float redPV[NW * 16 * 16];
    __shared__ int   redPI[NW * 16 * 16];
    __shared__ float redNV[NW * 16 * 16];
    __shared__ int   redNI[NW * 16 * 16];

    const int tid  = threadIdx.x;
    const int wave = tid >> 5;
    const int lane = tid & 31;
    const int l15  = lane & 15;
    const int hi   = lane >> 4;
    const int rowbase = blockIdx.x * 16;

    // Stage 16 x d A-panel straight into LDS with the CDNA5 async copy engine:
    // GLOBAL_LOAD_ASYNC_TO_LDS_B128 (no VGPR round-trip, tracked by ASYNCcnt).
    for (int idx = tid; idx < (16 * d) >> 2; idx += blockDim.x) {
        int e = idx << 2;
        int r = e / d;
        int k = e % d;
        __builtin_amdgcn_global_load_async_to_lds_b128(
            (gas_v4i*)&X[(size_t)(rowbase + r) * d + k],
            (las_v4i*)&As[r * AS_STRIDE + k],
            0, 0);
    }
    __builtin_amdgcn_s_wait_asynccnt(0);
    __syncthreads();

    // Per-lane row metadata: C-layout rows are M = r + 8*hi
    float sqRow[8]; int tgtRow[8];
#pragma unroll
    for (int r = 0; r < 8; ++r) {
        int m = r + hi * 8;
        sqRow[r]  = sq[rowbase + m];
        tgtRow[r] = tgt[rowbase + m];
    }

    float bPosV[8]; int bPosI[8]; float bNegV[8]; int bNegI[8];
#pragma unroll
    for (int r = 0; r < 8; ++r) {
        bPosV[r] = -INFINITY; bPosI[r] = -1;
        bNegV[r] =  INFINITY; bNegI[r] = -1;
    }

    // A-frag (16x4 f32): lanes 0-15 -> K={kk,kk+1}, lanes 16-31 -> K={kk+2,kk+3}
    const int koff = hi * 2;
    const float* AsBase = &As[l15 * AS_STRIDE + koff];

    const int ntiles = n >> 4;
    for (int t = wave; t < ntiles; t += NW) {
        const int colbase = t << 4;
        const int col = colbase + l15;           // C-layout: lane&15 == N
        const int   tgtCol = tgt[col];
        const float sqCol  = sq[col];
        const float* Bptr = X + (size_t)col * d + koff;   // B-frag mirrors A layout

        // Warm L0/L2 for the next tile's B column (global_prefetch_b8).
        {
            int tn = t + NW;
            if (tn < ntiles)
                __builtin_prefetch(X + (size_t)((tn << 4) + l15) * d + koff, 0, 0);
        }

        v8f acc = {};
#pragma unroll 8
        for (int kk = 0; kk < d; kk += 4) {
            v2f a = *(const v2f*)(AsBase + kk);          // ds_load (2addr-fused)
            v2f b = *(const v2f*)(Bptr   + kk);          // global_load_b64 (L2-resident)
            acc = __builtin_amdgcn_wmma_f32_16x16x4_f32(
                false, a, false, b, (short)0, acc, false, false);
        }

        // Epilogue: Gram -> distance -> index-aware running extrema
#pragma unroll
        for (int r = 0; r < 8; ++r) {
            float d2 = sqRow[r] + sqCol - 2.0f * acc[r];
            float dd = sqrtf(fmaxf(d2, EPS_D));
            if (tgtCol == tgtRow[r]) {
                if (dd > bPosV[r] || (dd == bPosV[r] && col > bPosI[r])) {
                    bPosV[r] = dd; bPosI[r] = col;
                }
            } else {
                if (dd < bNegV[r] || (dd == bNegV[r] && col > bNegI[r])) {
                    bNegV[r] = dd; bNegI[r] = col;
                }
            }
        }
    }

    // Publish per-lane candidates: [wave][rowInBlock][lane&15]
#pragma unroll
    for (int r = 0; r < 8; ++r) {
        int m = r + hi * 8;
        int idx = (wave * 16 + m) * 16 + l15;
        redPV[idx] = bPosV[r]; redPI[idx] = bPosI[r];
        redNV[idx] = bNegV[r]; redNI[idx] = bNegI[r];
    }
    __syncthreads();

    // Final combine: index-aware => exact torch last-tie semantics
    if (tid < 16) {
        int m = tid;
        float pv = -INFINITY; int pi = -1;
        float nv =  INFINITY; int ni = -1;
        for (int w = 0; w < NW; ++w)
            for (int l = 0; l < 16; ++l) {
                int idx = (w * 16 + m) * 16 + l;
                float cv = redPV[idx]; int ci = redPI[idx];
                if (cv > pv || (cv == pv && ci > pi)) { pv = cv; pi = ci; }
                float dv = redNV[idx]; int di = redNI[idx];
                if (dv < nv || (dv == nv && di > ni)) { nv = dv; ni = di; }
            }
        dist_ap[rowbase + m] = pv;
        dist_an[rowbase + m] = nv;
        jstar[rowbase + m] = pi;
        kstar[rowbase + m] = ni;
    }
}

// ---------------- Kernel C: dist(j*, k*) gather + per-row losses ----------------
__global__ void dan2_kernel(const float* __restrict__ X, const float* __restrict__ sq,
                            const float* __restrict__ dist_ap, const float* __restrict__ dist_an,
                            const int* __restrict__ jstar, const int* __restrict__ kstar,
                            float* __restrict__ loss, float* __restrict__ loss3,
                            int n, int d) {
    int wave = threadIdx.x >> 5;
    int lane = threadIdx.x & 31;
    int i = blockIdx.x * (blockDim.x >> 5) + wave;
    if (i >= n) return;
    int j = jstar[i], k = kstar[i];
    const float* xj = X + (size_t)j * d;
    const float* xk = X + (size_t)k * d;
    float s = 0.f;
    for (int t = lane; t < d; t += 32) s += xj[t] * xk[t];
    for (int off = 16; off > 0; off >>= 1) s += __shfl_down(s, off, 32);
    if (lane == 0) {
        float d2   = sq[j] + sq[k] - 2.0f * s;
        float dan2 = sqrtf(fmaxf(d2, EPS_D));
        float ap = dist_ap[i], an = dist_an[i];
        loss[i]  = fmaxf(ap - an, 0.0f);        // MARGIN = 0
        loss3[i] = fabsf(an - dan2);
    }
}

// ---------------- Kernel D: deterministic final reduction ----------------
__global__ void finalize_kernel(const float* __restrict__ loss,
                                const float* __restrict__ loss3,
                                float* __restrict__ out, int n) {
    __shared__ float s1[256], s2[256];
    int tid = threadIdx.x;
    float a = 0.f, b = 0.f;
    for (int i = tid; i < n; i += 256) { a += loss[i]; b += loss3[i]; }
    s1[tid] = a; s2[tid] = b;
    __syncthreads();
    for (int off = 128; off > 0; off >>= 1) {
        if (tid < off) { s1[tid] += s1[tid + off]; s2[tid] += s2[tid + off]; }
        __syncthreads();
    }
    if (tid == 0) {
        float l1 = s1[0] / n, l2 = s2[0] / n;
        out[0] = l1 + l2;   // loss_final
        out[1] = l1;        // l1
        out[2] = l2;        // l2
    }
}

// ---------------- Launch ----------------
extern "C" void kernel_launch(void* const* d_in, const int* in_sizes, int n_in,
                              void* d_out, int out_size, void* d_ws, size_t ws_size,
                              hipStream_t stream) {
    const float* X  = (const float*)d_in[0];
    const int* tgt  = (const int*)d_in[1];
    const int n = in_sizes[1];
    const int d = in_sizes[0] / n;
    float* out = (float*)d_out;

    float* ws      = (float*)d_ws;
    float* sq      = ws;
    float* dist_ap = ws + (size_t)n;
    float* dist_an = ws + (size_t)2 * n;
    int*   jstar   = (int*)(ws + (size_t)3 * n);
    int*   kstar   = (int*)(ws + (size_t)4 * n);
    float* loss    = ws + (size_t)5 * n;
    float* loss3   = ws + (size_t)6 * n;

    rownorm_kernel<<<(n + 7) / 8, 256, 0, stream>>>(X, sq, n, d);
    hardmine_kernel<<<n / 16, NW * 32, 0, stream>>>(X, tgt, sq, dist_ap, dist_an,
                                                    jstar, kstar, n, d);
    dan2_kernel<<<(n + 7) / 8, 256, 0, stream>>>(X, sq, dist_ap, dist_an,
                                                 jstar, kstar, loss, loss3, n, d);
    finalize_kernel<<<1, 256, 0, stream>>>(loss, loss3, out, n);
}